// L3_77206332113743
// MI455X (gfx1250) — compile-verified
//
#include <hip/hip_runtime.h>
#include <hip/hip_bf16.h>

typedef __attribute__((ext_vector_type(2))) float v2f;
typedef __attribute__((ext_vector_type(8))) float v8f;

#define N_NODES 100000   // divisible by 16 -> no M-tail in WMMA tiles
#define N_EDGES 3200000

// ---------------------------------------------------------------------------
// x = concat(one_hot[N,20], features[N,44]) -> [N,64]
// ---------------------------------------------------------------------------
__global__ void concat64_kernel(const float* __restrict__ oh,
                                const float* __restrict__ ft,
                                float* __restrict__ x) {
  int idx = blockIdx.x * blockDim.x + threadIdx.x;
  if (idx >= N_NODES * 64) return;
  int n = idx >> 6, j = idx & 63;
  x[idx] = (j < 20) ? oh[n * 20 + j] : ft[n * 44 + (j - 20)];
}

// ---------------------------------------------------------------------------
// Wt[r][c][k] = (c < out) ? W[r][k][c] : 0   -- zero-padded transpose so the
// WMMA B fragment {W[ka][col], W[ka+1][col]} is one contiguous 8-byte load.
// ---------------------------------------------------------------------------
__global__ void wpad_transpose_kernel(const float* __restrict__ W,  // [4][in][out]
                                      float* __restrict__ Wt,       // [4][out_pad][in]
                                      int in, int out, int out_pad) {
  int idx = blockIdx.x * blockDim.x + threadIdx.x;
  int total = 4 * out_pad * in;
  if (idx >= total) return;
  int k = idx % in;
  int c = (idx / in) % out_pad;
  int r = idx / (in * out_pad);
  Wt[idx] = (c < out) ? W[(size_t)r * in * out + (size_t)k * out + c] : 0.f;
}

// ---------------------------------------------------------------------------
// z[n, r, :] = x[n, :IN] @ W[r]   via V_WMMA_F32_16X16X4_F32.
// One wave per (16-row tile, channel r, 16-col tile). Fully unrolled K-loop,
// unpredicated b64 loads for both fragments, unconditional stores.
// A layout (32-bit 16x4): lane<16: row=lane%16, K={k0,k0+1};
//                         lane>=16: K={k0+2,k0+3}.
// D layout: VGPR i -> M=i (lanes 0-15) / M=8+i (lanes 16-31), N=lane%16.
// ---------------------------------------------------------------------------
template <int IN, int OUT_PAD>
__global__ void gconv_transform_wmma(const float* __restrict__ x,
                                     const float* __restrict__ Wt,  // [4][OUT_PAD][IN]
                                     float* __restrict__ z) {       // [N][4][OUT_PAD]
  constexpr int NTN = OUT_PAD / 16;
  constexpr int TILES_M = N_NODES / 16;
  int wave = (blockIdx.x * blockDim.x + threadIdx.x) >> 5;
  int lane = threadIdx.x & 31;
  if (wave >= TILES_M * 4 * NTN) return;   // wave-uniform

  int tn = wave % NTN;
  int r  = (wave / NTN) & 3;
  int tm = wave / (NTN * 4);

  int row = tm * 16 + (lane & 15);
  int klo = (lane >> 4) << 1;              // 0 for lanes 0-15, 2 for 16-31
  int col = tn * 16 + (lane & 15);

  const float* ap = x  + (size_t)row * IN + klo;
  const float* bp = Wt + (size_t)(r * OUT_PAD + col) * IN + klo;

  v8f c = {};
#pragma unroll
  for (int t = 0; t < IN / 4; ++t) {
    v2f a = *(const v2f*)(ap + 4 * t);
    v2f b = *(const v2f*)(bp + 4 * t);
    c = __builtin_amdgcn_wmma_f32_16x16x4_f32(
            /*neg_a=*/false, a, /*neg_b=*/false, b,
            /*c_mod=*/(short)0, c, /*reuse_a=*/false, /*reuse_b=*/false);
  }

  int mbase = tm * 16 + ((lane >> 4) << 3);
  int nn = tn * 16 + (lane & 15);
  float* zp = z + (size_t)mbase * (4 * OUT_PAD) + r * OUT_PAD + nn;
#pragma unroll
  for (int i = 0; i < 8; ++i)
    zp[(size_t)i * 4 * OUT_PAD] = c[i];
}

// ---------------------------------------------------------------------------
// Edge phase: acc[dst] += sum_r ew[r][e] * z[src][r][:]
// One thread per edge; b128 gathers from L2-resident z; native f32 atomics
// into L2-resident acc.  OUT4 = out/4 (all layer outs are multiples of 4).
// ---------------------------------------------------------------------------
template <int OUT4>
__global__ void gconv_edge_kernel(const float* __restrict__ z,
                                  const int* __restrict__ src,
                                  const int* __restrict__ dst,
                                  const float* __restrict__ ew,
                                  float* __restrict__ acc,
                                  int out_pad) {
  int e = blockIdx.x * blockDim.x + threadIdx.x;
  if (e >= N_EDGES) return;
  int s = src[e], d = dst[e];
  float w0 = ew[e];
  float w1 = ew[N_EDGES + e];
  float w2 = ew[2 * N_EDGES + e];
  float w3 = ew[3 * N_EDGES + e];
  int op4 = out_pad >> 2;
  const float4* z0 = (const float4*)(z + (size_t)s * 4 * out_pad);
  const float4* z1 = z0 + op4;
  const float4* z2 = z1 + op4;
  const float4* z3 = z2 + op4;
  float* ap = acc + (size_t)d * (OUT4 * 4);
#pragma unroll
  for (int j = 0; j < OUT4; ++j) {
    float4 a = z0[j], b = z1[j], cc = z2[j], dd = z3[j];
    float vx = w0 * a.x + w1 * b.x + w2 * cc.x + w3 * dd.x;
    float vy = w0 * a.y + w1 * b.y + w2 * cc.y + w3 * dd.y;
    float vz = w0 * a.z + w1 * b.z + w2 * cc.z + w3 * dd.z;
    float vw = w0 * a.w + w1 * b.w + w2 * cc.w + w3 * dd.w;
    unsafeAtomicAdd(ap + 4 * j + 0, vx);
    unsafeAtomicAdd(ap + 4 * j + 1, vy);
    unsafeAtomicAdd(ap + 4 * j + 2, vz);
    unsafeAtomicAdd(ap + 4 * j + 3, vw);
  }
}

// ---------------------------------------------------------------------------
// x_next = elu(acc + bias)
// ---------------------------------------------------------------------------
__global__ void bias_elu_kernel(const float* __restrict__ acc,
                                const float* __restrict__ b,
                                float* __restrict__ xo, int out) {
  int idx = blockIdx.x * blockDim.x + threadIdx.x;
  if (idx >= N_NODES * out) return;
  int j = idx % out;
  float v = acc[idx] + b[j];
  xo[idx] = v > 0.f ? v : (__expf(v) - 1.f);
}

// ---------------------------------------------------------------------------
// Head MLP: 4 -> 8 -> 4 -> 1 (elu, elu, sigmoid), one thread per node.
// ---------------------------------------------------------------------------
__global__ void head_mlp_kernel(const float* __restrict__ x,
                                const float* __restrict__ lw1,
                                const float* __restrict__ lb1,
                                const float* __restrict__ lw2,
                                const float* __restrict__ lb2,
                                const float* __restrict__ lw3,
                                const float* __restrict__ lb3,
                                float* __restrict__ out) {
  int n = blockIdx.x * blockDim.x + threadIdx.x;
  if (n >= N_NODES) return;
  float h0[4];
#pragma unroll
  for (int i = 0; i < 4; ++i) h0[i] = x[n * 4 + i];
  float h1[8];
#pragma unroll
  for (int j = 0; j < 8; ++j) {
    float s = lb1[j];
#pragma unroll
    for (int i = 0; i < 4; ++i) s += h0[i] * lw1[i * 8 + j];
    h1[j] = s > 0.f ? s : (__expf(s) - 1.f);
  }
  float h2[4];
#pragma unroll
  for (int j = 0; j < 4; ++j) {
    float s = lb2[j];
#pragma unroll
    for (int i = 0; i < 8; ++i) s += h1[i] * lw2[i * 4 + j];
    h2[j] = s > 0.f ? s : (__expf(s) - 1.f);
  }
  float s = lb3[0];
#pragma unroll
  for (int i = 0; i < 4; ++i) s += h2[i] * lw3[i];
  out[n] = 1.f / (1.f + __expf(-s));
}

// ---------------------------------------------------------------------------
extern "C" void kernel_launch(void* const* d_in, const int* in_sizes, int n_in,
                              void* d_out, int out_size, void* d_ws,
                              size_t ws_size, hipStream_t stream) {
  const float* one_hot  = (const float*)d_in[0];
  const float* features = (const float*)d_in[1];
  // d_in[2] gemme_features: unused by the reference
  const int*   esrc = (const int*)d_in[3];
  const int*   edst = (const int*)d_in[4];
  const float* ew   = (const float*)d_in[5];
  const float* W[5] = {(const float*)d_in[6],  (const float*)d_in[8],
                       (const float*)d_in[10], (const float*)d_in[12],
                       (const float*)d_in[14]};
  const float* B[5] = {(const float*)d_in[7],  (const float*)d_in[9],
                       (const float*)d_in[11], (const float*)d_in[13],
                       (const float*)d_in[15]};
  const float* lw1 = (const float*)d_in[16];
  const float* lb1 = (const float*)d_in[17];
  const float* lw2 = (const float*)d_in[18];
  const float* lb2 = (const float*)d_in[19];
  const float* lw3 = (const float*)d_in[20];
  const float* lb3 = (const float*)d_in[21];

  // Workspace (floats): xA[N*64] | xB[N*32] | z[N*4*32] | acc[N*32] | Wt[8192]
  float* ws  = (float*)d_ws;
  float* xA  = ws;
  float* xB  = ws + (size_t)N_NODES * 64;
  float* z   = ws + (size_t)N_NODES * 96;
  float* acc = ws + (size_t)N_NODES * 224;
  float* Wt  = ws + (size_t)N_NODES * 256;

  concat64_kernel<<<(N_NODES * 64 + 255) / 256, 256, 0, stream>>>(one_hot,
                                                                  features, xA);

  const int ins[5]  = {64, 20, 16, 12, 8};
  const int outs[5] = {20, 16, 12, 8, 4};
  float* xcur = xA;
  float* xnxt = xB;

  for (int l = 0; l < 5; ++l) {
    int in = ins[l], out = outs[l];
    int out_pad = (out > 16) ? 32 : 16;
    int ntn = out_pad / 16;

    int wtN = 4 * out_pad * in;
    wpad_transpose_kernel<<<(wtN + 255) / 256, 256, 0, stream>>>(W[l], Wt, in,
                                                                 out, out_pad);

    long thr = (long)(N_NODES / 16) * 4 * ntn * 32;
    int tb = (int)((thr + 255) / 256);
    switch (l) {
      case 0: gconv_transform_wmma<64, 32><<<tb, 256, 0, stream>>>(xcur, Wt, z); break;
      case 1: gconv_transform_wmma<20, 16><<<tb, 256, 0, stream>>>(xcur, Wt, z); break;
      case 2: gconv_transform_wmma<16, 16><<<tb, 256, 0, stream>>>(xcur, Wt, z); break;
      case 3: gconv_transform_wmma<12, 16><<<tb, 256, 0, stream>>>(xcur, Wt, z); break;
      default: gconv_transform_wmma<8, 16><<<tb, 256, 0, stream>>>(xcur, Wt, z); break;
    }

    hipMemsetAsync(acc, 0, (size_t)N_NODES * out * sizeof(float), stream);

    int blocksE = (N_EDGES + 255) / 256;
    switch (out / 4) {
      case 5: gconv_edge_kernel<5><<<blocksE, 256, 0, stream>>>(z, esrc, edst, ew, acc, out_pad); break;
      case 4: gconv_edge_kernel<4><<<blocksE, 256, 0, stream>>>(z, esrc, edst, ew, acc, out_pad); break;
      case 3: gconv_edge_kernel<3><<<blocksE, 256, 0, stream>>>(z, esrc, edst, ew, acc, out_pad); break;
      case 2: gconv_edge_kernel<2><<<blocksE, 256, 0, stream>>>(z, esrc, edst, ew, acc, out_pad); break;
      default: gconv_edge_kernel<1><<<blocksE, 256, 0, stream>>>(z, esrc, edst, ew, acc, out_pad); break;
    }

    bias_elu_kernel<<<(N_NODES * out + 255) / 256, 256, 0, stream>>>(acc, B[l],
                                                                     xnxt, out);
    float* t = xcur; xcur = xnxt; xnxt = t;
  }

  head_mlp_kernel<<<(N_NODES + 255) / 256, 256, 0, stream>>>(
      xcur, lw1, lb1, lw2, lb2, lw3, lb3, (float*)d_out);
}